// CurrLearnLoss_32203664785639
// MI455X (gfx1250) — compile-verified
//
#include <hip/hip_runtime.h>
#include <hip/hip_bf16.h>
#include <stdint.h>

#define S_DIM 120
#define B_DIM 32768
#define C_DIM 7
#define P_DIM 9
#define NBLK2 1920            // 1920 * 256 * 8 == S*B exactly
#define EPT2  8               // elements per thread in phase 2

typedef __attribute__((ext_vector_type(2))) float v2f;
typedef __attribute__((ext_vector_type(8))) float v8f;

// ---------------------------------------------------------------------------
// Kernel 0: high-occupancy pack. One thread per (s,b): argmax over P=9 qmask
// values (first-max tie-break like jnp.argmax), pack speaker id (4b), label
// (3b) and umask bit (1b) into one byte. qmask/label/umask are streamed with
// non-temporal loads (no reuse -> keep the 192MB L2 for packed/curr_w/prob
// working set). S*B threads = 122880 waves: full latency hiding for the
// 173MB stream.
// ---------------------------------------------------------------------------
__global__ void __launch_bounds__(256)
k_pack(const float* __restrict__ qmask,
       const int*   __restrict__ label,
       const float* __restrict__ umask,
       unsigned char* __restrict__ packed) {
  const int idx = blockIdx.x * 256 + threadIdx.x;      // exact cover of S*B
  const float* q = qmask + (size_t)idx * P_DIM;

  float best = __builtin_nontemporal_load(&q[0]);
  int qm = 0;
#pragma unroll
  for (int p = 1; p < P_DIM; ++p) {
    float v = __builtin_nontemporal_load(&q[p]);
    if (v > best) { best = v; qm = p; }
  }
  const int   lab = __builtin_nontemporal_load(&label[idx]);
  const float um  = __builtin_nontemporal_load(&umask[idx]);
  const unsigned char byte =
      (unsigned char)(qm | (lab << 4) | ((um == 1.0f) ? 0x80 : 0));
  packed[idx] = byte;                                  // RT: stays in L2
}

// ---------------------------------------------------------------------------
// Kernel 1: per-batch-column speaker recurrence -> curr_w[b].
// Block = 256 consecutive b columns. The 120x256-byte packed tile is staged
// into LDS with global_load_async_to_lds_b64 (wave w stages rows {k*8+w},
// 8B per lane = one full 256B row per wave per op), one s_wait_asynccnt 0 +
// barrier, then the sequential S recurrence runs entirely from LDS with all
// P=9 per-speaker state in registers (unrolled predicated updates).
// last_lab init -1 encodes "never spoke" exactly like the reference's
// (conv == -1) term; state updates regardless of umask, matching reference.
// ---------------------------------------------------------------------------
__global__ void __launch_bounds__(256)
k_weights(const unsigned char* __restrict__ packed,
          const int*   __restrict__ iters,
          float*       __restrict__ curr_w) {
  __shared__ unsigned char tile[S_DIM * 256];          // 30720 B

  const int t  = threadIdx.x;
  const int b0 = blockIdx.x * 256;

  const uint32_t lds0  = (uint32_t)(uintptr_t)tile;    // LDS byte offset
  const uint64_t gbase = (uint64_t)(packed + b0);

  // Stage 120 rows of 256 bytes: 15 async ops, 8 rows per op.
  const int r_in_op = t >> 5;                          // 0..7 (wave id)
  const int col     = (t & 31) * 8;                    // 0..248, 8B aligned
#pragma unroll
  for (int k = 0; k < 15; ++k) {
    const int r = k * 8 + r_in_op;                     // row 0..119
    const uint32_t voff  = (uint32_t)(r * B_DIM + col);
    const uint32_t laddr = lds0 + (uint32_t)(r * 256 + col);
    asm volatile("global_load_async_to_lds_b64 %0, %1, %2"
                 :
                 : "v"(laddr), "v"(voff), "s"(gbase)
                 : "memory");
  }
  asm volatile("s_wait_asynccnt 0" ::: "memory");
  __syncthreads();

  float pes[P_DIM], pel[P_DIM];
  int   last_lab[P_DIM];
#pragma unroll
  for (int p = 0; p < P_DIM; ++p) {
    pes[p] = 0.0f;
    pel[p] = 1e-5f;
    last_lab[p] = -1;
  }

  for (int i = 0; i < S_DIM; ++i) {
    const unsigned int v = tile[i * 256 + t];
    const int   qm     = v & 0xF;
    const int   lab    = (v >> 4) & 7;
    const float active = (v & 0x80) ? 1.0f : 0.0f;
#pragma unroll
    for (int p = 0; p < P_DIM; ++p) {
      if (p == qm) {
        const float shift = (lab != last_lab[p]) ? 1.0f : 0.0f;
        pes[p] += shift * active;
        pel[p] += active;
        last_lab[p] = lab;
      }
    }
  }

  float n_spk = 0.0f;
#pragma unroll
  for (int p = 0; p < P_DIM; ++p) n_spk += (pel[p] >= 1.0f) ? 1.0f : 0.0f;
  float wd = 0.0f;
#pragma unroll
  for (int p = 0; p < P_DIM; ++p) wd += pes[p] / pel[p] / n_spk;

  const float mu = (float)iters[0] / 6.0f;
  const float x  = (mu - wd) / 0.1f;
  curr_w[b0 + t] = 1.0f / (1.0f + expf(-x));
}

// ---------------------------------------------------------------------------
// Kernel 2: partial sums of prob[s,b,label]*umask*curr_w. label/umask come
// from the packed byte array (3.9MB, L2-resident) instead of 31MB of raw
// arrays. The prob gather goes through the CDNA5 async path with a
// non-temporal hint (prob is touched once; keep it out of L2): 8 gathers
// per lane into private LDS slots, one s_wait_asynccnt 0.
// ---------------------------------------------------------------------------
__global__ void __launch_bounds__(256)
k_nll(const float* __restrict__ prob,
      const unsigned char* __restrict__ packed,
      const float* __restrict__ curr_w,
      float*       __restrict__ partials) {
  __shared__ float stage[EPT2][256];
  __shared__ float red[256];

  const int t      = threadIdx.x;
  const int base   = blockIdx.x * 256 + t;             // < 491520
  const int stride = NBLK2 * 256;                      // 491520

  const uint32_t lds0  = (uint32_t)(uintptr_t)&stage[0][t];
  const uint64_t pbase = (uint64_t)prob;

  float w_r[EPT2];
#pragma unroll
  for (int e = 0; e < EPT2; ++e) {
    const int idx = base + e * stride;                 // s*B + b, exact cover
    const unsigned int v = packed[idx];
    const int   lab = (v >> 4) & 7;
    const float um  = (v & 0x80) ? 1.0f : 0.0f;
    w_r[e] = um * curr_w[idx & (B_DIM - 1)];
    const uint32_t poff  = (uint32_t)((idx * C_DIM + lab) * 4);
    const uint32_t laddr = lds0 + (uint32_t)(e * 256 * 4);
    asm volatile("global_load_async_to_lds_b32 %0, %1, %2 th:TH_LOAD_NT"
                 :
                 : "v"(laddr), "v"(poff), "s"(pbase)
                 : "memory");
  }
  asm volatile("s_wait_asynccnt 0" ::: "memory");

  float acc = 0.0f;
#pragma unroll
  for (int e = 0; e < EPT2; ++e) acc += stage[e][t] * w_r[e];

  red[t] = acc;
  __syncthreads();
#pragma unroll
  for (int off = 128; off > 0; off >>= 1) {
    if (t < off) red[t] += red[t + off];
    __syncthreads();
  }
  if (t == 0) partials[blockIdx.x] = red[0];
}

// ---------------------------------------------------------------------------
// Kernel 3: one wave32. Each lane serially sums 60 partials (deterministic),
// then the 32 lane values are reduced with V_WMMA_F32_16X16X4_F32:
// A[16x4] holds one lane value per slot (other VGPR = 0), B[4x16] = ones,
// so D[m][n] = sum_k A[m][k]; lane 0 holds D[0..7][0], lane 16 D[8..15][0].
// ---------------------------------------------------------------------------
__global__ void __launch_bounds__(32)
k_final(const float* __restrict__ partials, float* __restrict__ out) {
  const int t = threadIdx.x;
  float x = 0.0f;
  for (int j = 0; j < NBLK2 / 32; ++j) x += partials[j * 32 + t];

  v2f a;  a.x = x;     a.y = 0.0f;   // each lane fills exactly one A slot
  v2f bm; bm.x = 1.0f; bm.y = 1.0f;  // B = all ones (layout-independent)
  v8f c = {};
  c = __builtin_amdgcn_wmma_f32_16x16x4_f32(
      /*neg_a=*/false, a, /*neg_b=*/false, bm,
      /*c_mod=*/(short)0, c, /*reuse_a=*/false, /*reuse_b=*/false);

  float rowsum = c[0] + c[1] + c[2] + c[3] + c[4] + c[5] + c[6] + c[7];
  float other  = __shfl(rowsum, 16, 32);
  if (t == 0) {
    out[0] = -(rowsum + other) / (float)(S_DIM * B_DIM);
  }
}

// ---------------------------------------------------------------------------
extern "C" void kernel_launch(void* const* d_in, const int* in_sizes, int n_in,
                              void* d_out, int out_size, void* d_ws, size_t ws_size,
                              hipStream_t stream) {
  const float* prob  = (const float*)d_in[0];   // [S,B,C] f32
  const int*   label = (const int*)  d_in[1];   // [S,B]   i32
  const float* umask = (const float*)d_in[2];   // [S,B]   f32
  const float* qmask = (const float*)d_in[3];   // [S,B,P] f32
  const int*   iters = (const int*)  d_in[4];   // scalar

  float*         curr_w   = (float*)d_ws;                      // B floats
  float*         partials = curr_w + B_DIM;                    // NBLK2 floats
  unsigned char* packed   = (unsigned char*)(partials + NBLK2);// S*B bytes, 8B-aligned
  float*         out      = (float*)d_out;

  k_pack   <<<(S_DIM * B_DIM) / 256, 256, 0, stream>>>(qmask, label, umask, packed);
  k_weights<<<B_DIM / 256,           256, 0, stream>>>(packed, iters, curr_w);
  k_nll    <<<NBLK2,                 256, 0, stream>>>(prob, packed, curr_w, partials);
  k_final  <<<1,                      32, 0, stream>>>(partials, out);
}